// GraphAE_687194767905
// MI455X (gfx1250) — compile-verified
//
#include <hip/hip_runtime.h>

#define NNODES 50000
#define NEDGES 800000

typedef __attribute__((ext_vector_type(2))) float v2f;
typedef __attribute__((ext_vector_type(8))) float v8f;

// ---------------- utility kernels ----------------

__global__ void zero_f32(float* __restrict__ p, int n) {
  int i = blockIdx.x * blockDim.x + threadIdx.x;
  if (i < n) p[i] = 0.0f;
}

__global__ void deg_kernel(const long long* __restrict__ dst,
                           float* __restrict__ deg, int E) {
  int e = blockIdx.x * blockDim.x + threadIdx.x;
  if (e < E) atomicAdd(&deg[(int)dst[e]], 1.0f);
}

__global__ void invdeg_kernel(float* __restrict__ deg, int n) {
  int i = blockIdx.x * blockDim.x + threadIdx.x;
  if (i < n) deg[i] = 1.0f / fmaxf(deg[i], 1.0f);
}

// scatter-add of source features into destination accumulator.
// One thread per (edge, feature). F is a power of two -> cheap index math.
// All traffic is L2-resident on MI455X (192MB L2 >> 25.6MB working set).
template <int F>
__global__ void scatter_kernel(const long long* __restrict__ src,
                               const long long* __restrict__ dst,
                               const float* __restrict__ x,
                               float* __restrict__ agg, int E) {
  int g = blockIdx.x * blockDim.x + threadIdx.x;
  int e = g / F;            // becomes a shift (F constexpr pow2)
  int f = g & (F - 1);
  if (e < E) {
    int s = (int)src[e];
    int d = (int)dst[e];
    atomicAdd(&agg[(size_t)d * F + f], x[(size_t)s * F + f]);
  }
}

// ---------------- fused mean-normalize + GEMM + epilogue ----------------
// Block: 16 node rows, 2*FOUT threads (FOUT/16 waves, one 16x16 tile each).
// Uses exact f32 WMMA: V_WMMA_F32_16X16X4_F32 (D = A(16x4) * B(4x16) + C).
// MODE 0: L2-normalize rows then ReLU.  MODE 1: row softmax.
template <int FIN, int FOUT, int MODE>
__global__ __launch_bounds__(2 * FOUT)
void sage_gemm(const float* __restrict__ agg,
               const float* __restrict__ invdeg,
               const float* __restrict__ W,
               const float* __restrict__ bias,
               float* __restrict__ out) {
  constexpr int SA = FIN + 4;    // padded LDS strides: lanes step by SA -> banks
  constexpr int SO = FOUT + 4;   // differ by 4 mod 64 -> conflict free
  __shared__ float smA[16 * SA];
  __shared__ float smO[16 * SO];
  __shared__ float smR[16];
  __shared__ float smR2[16];

  const int tid = threadIdx.x;
  const int m0  = blockIdx.x * 16;

  // stage deg-normalized mean tile [16 x FIN] into LDS
  for (int idx = tid; idx < 16 * FIN; idx += blockDim.x) {
    int row = idx / FIN;
    int k   = idx - row * FIN;
    int node = m0 + row;
    smA[row * SA + k] = agg[(size_t)node * FIN + k] * invdeg[node];
  }
  __syncthreads();

  const int lane = tid & 31;
  const int l15  = lane & 15;
  const int hi   = lane >> 4;         // 0: K pair {0,1}, 1: K pair {2,3}
  const int n0   = (tid >> 5) * 16;   // this wave's output-column tile

  v8f c = {0.f, 0.f, 0.f, 0.f, 0.f, 0.f, 0.f, 0.f};
  for (int k = 0; k < FIN; k += 4) {
    const int ka = k + 2 * hi;
    v2f a, b;
    // A 16x4 f32 layout: v0 = K=ka, v1 = K=ka+1, row M = l15
    a.x = smA[l15 * SA + ka];
    a.y = smA[l15 * SA + ka + 1];
    // B 4x16 f32 layout (mirror): v0 = K=ka, v1 = K=ka+1, col N = n0+l15
    b.x = W[(size_t)ka * FOUT + n0 + l15];
    b.y = W[(size_t)(ka + 1) * FOUT + n0 + l15];
    c = __builtin_amdgcn_wmma_f32_16x16x4_f32(
        /*neg_a=*/false, a, /*neg_b=*/false, b,
        /*c_mod=*/(short)0, c, /*reuse_a=*/false, /*reuse_b=*/false);
  }

  // bias + spill 16x16 tile to LDS (C/D layout: VGPR r -> M=r (lanes 0-15),
  // M=r+8 (lanes 16-31); N = lane&15)
  const float bv = bias[n0 + l15];
#pragma unroll
  for (int r = 0; r < 8; ++r)
    smO[(r + 8 * hi) * SO + n0 + l15] = c[r] + bv;
  __syncthreads();

  if (MODE == 0) {
    // row L2 norm (16 threads, FOUT<=128 -> trivial serial reduction)
    if (tid < 16) {
      float ss = 0.0f;
      for (int j = 0; j < FOUT; ++j) {
        float v = smO[tid * SO + j];
        ss += v * v;
      }
      smR[tid] = 1.0f / fmaxf(sqrtf(ss), 1e-12f);
    }
    __syncthreads();
    for (int idx = tid; idx < 16 * FOUT; idx += blockDim.x) {
      int row = idx / FOUT;
      int col = idx - row * FOUT;
      out[(size_t)(m0 + row) * FOUT + col] =
          fmaxf(smO[row * SO + col] * smR[row], 0.0f);
    }
  } else {
    // row softmax
    if (tid < 16) {
      float m = -3.402823466e38f;
      for (int j = 0; j < FOUT; ++j) m = fmaxf(m, smO[tid * SO + j]);
      float s = 0.0f;
      for (int j = 0; j < FOUT; ++j) s += __expf(smO[tid * SO + j] - m);
      smR[tid]  = m;
      smR2[tid] = 1.0f / s;
    }
    __syncthreads();
    for (int idx = tid; idx < 16 * FOUT; idx += blockDim.x) {
      int row = idx / FOUT;
      int col = idx - row * FOUT;
      out[(size_t)(m0 + row) * FOUT + col] =
          __expf(smO[row * SO + col] - smR[row]) * smR2[row];
    }
  }
}

// ---------------- driver ----------------

extern "C" void kernel_launch(void* const* d_in, const int* in_sizes, int n_in,
                              void* d_out, int out_size, void* d_ws, size_t ws_size,
                              hipStream_t stream) {
  const float*     x    = (const float*)d_in[0];
  const long long* eidx = (const long long*)d_in[1];  // int64 [2, E]
  const float* W1 = (const float*)d_in[2];
  const float* b1 = (const float*)d_in[3];
  const float* W2 = (const float*)d_in[4];
  const float* b2 = (const float*)d_in[5];
  const float* W3 = (const float*)d_in[6];
  const float* b3 = (const float*)d_in[7];
  const float* W4 = (const float*)d_in[8];
  const float* b4 = (const float*)d_in[9];

  const long long* src = eidx;
  const long long* dst = eidx + NEDGES;

  float* ws  = (float*)d_ws;
  float* deg = ws;                               // N floats (becomes invdeg)
  float* agg = deg + NNODES;                     // N*128 floats
  float* hA  = agg + (size_t)NNODES * 128;       // N*128 floats
  float* hB  = hA  + (size_t)NNODES * 128;       // N*128 floats
  float* out = (float*)d_out;                    // [N, 32]

  const int T = 256;
  const int gN    = (NNODES + T - 1) / T;
  const int gE    = (NEDGES + T - 1) / T;
  const int gN64  = (NNODES * 64  + T - 1) / T;
  const int gN128 = (NNODES * 128 + T - 1) / T;
  const int gE64  = (NEDGES * 64  + T - 1) / T;   // 51.2M threads
  const int gE128 = (NEDGES * 128 + T - 1) / T;   // 102.4M threads
  const int gM    = NNODES / 16;                  // 3125 row-tile blocks

  // degree (once, reused by every layer), then invert in place
  zero_f32<<<gN, T, 0, stream>>>(deg, NNODES);
  deg_kernel<<<gE, T, 0, stream>>>(dst, deg, NEDGES);
  invdeg_kernel<<<gN, T, 0, stream>>>(deg, NNODES);

  // layer 1: x[64] -> hA[128], normalize+relu
  zero_f32<<<gN64, T, 0, stream>>>(agg, NNODES * 64);
  scatter_kernel<64><<<gE64, T, 0, stream>>>(src, dst, x, agg, NEDGES);
  sage_gemm<64, 128, 0><<<gM, 256, 0, stream>>>(agg, deg, W1, b1, hA);

  // layer 2: hA[128] -> hB[64], normalize+relu
  zero_f32<<<gN128, T, 0, stream>>>(agg, NNODES * 128);
  scatter_kernel<128><<<gE128, T, 0, stream>>>(src, dst, hA, agg, NEDGES);
  sage_gemm<128, 64, 0><<<gM, 128, 0, stream>>>(agg, deg, W2, b2, hB);

  // layer 3: hB[64] -> hA[128], normalize+relu
  zero_f32<<<gN64, T, 0, stream>>>(agg, NNODES * 64);
  scatter_kernel<64><<<gE64, T, 0, stream>>>(src, dst, hB, agg, NEDGES);
  sage_gemm<64, 128, 0><<<gM, 256, 0, stream>>>(agg, deg, W3, b3, hA);

  // layer 4: hA[128] -> out[32], softmax
  zero_f32<<<gN128, T, 0, stream>>>(agg, NNODES * 128);
  scatter_kernel<128><<<gE128, T, 0, stream>>>(src, dst, hA, agg, NEDGES);
  sage_gemm<128, 32, 1><<<gM, 64, 0, stream>>>(agg, deg, W4, b4, out);
}